// OneLayerSGC_20710332301834
// MI455X (gfx1250) — compile-verified
//
#include <hip/hip_runtime.h>
#include <hip/hip_bf16.h>

#define IN_F  128
#define OUT_F 64

typedef float v2f __attribute__((ext_vector_type(2)));
typedef float v8f __attribute__((ext_vector_type(8)));

// ---------------------------------------------------------------------------
// deg init
__global__ void zero_f32(float* __restrict__ p, int n) {
    int i = blockIdx.x * blockDim.x + threadIdx.x;
    if (i < n) p[i] = 0.0f;
}

// deg[dst[e]] += 1  (in-degrees)
__global__ void degree_kernel(const int* __restrict__ dst, float* __restrict__ deg, int nE) {
    int e = blockIdx.x * blockDim.x + threadIdx.x;
    if (e < nE) unsafeAtomicAdd(&deg[dst[e]], 1.0f);
}

// norm = clamp(deg,1)^-0.5  (in place)
__global__ void norm_kernel(float* __restrict__ norm, int n) {
    int i = blockIdx.x * blockDim.x + threadIdx.x;
    if (i < n) norm[i] = rsqrtf(fmaxf(norm[i], 1.0f));
}

// h0 = x * norm[row] ; h1 = 0
__global__ void prescale_kernel(const float* __restrict__ x, const float* __restrict__ norm,
                                float* __restrict__ h0, float* __restrict__ h1, int total) {
    int i = blockIdx.x * blockDim.x + threadIdx.x;
    if (i < total) {
        h0[i] = x[i] * norm[i >> 7];   // IN_F == 128
        h1[i] = 0.0f;
    }
}

// h0 = h1 * norm[row]^2 ; h1 = 0   (combines dst-scale of hop1 and src-scale of hop2)
__global__ void midscale_kernel(const float* __restrict__ norm, float* __restrict__ h0,
                                float* __restrict__ h1, int total) {
    int i = blockIdx.x * blockDim.x + threadIdx.x;
    if (i < total) {
        float nv = norm[i >> 7];
        h0[i] = h1[i] * nv * nv;
        h1[i] = 0.0f;
    }
}

// h0 = h1 * norm[row]   (final dst-scale, feeds the GEMM)
__global__ void finalscale_kernel(const float* __restrict__ norm, float* __restrict__ h0,
                                  const float* __restrict__ h1, int total) {
    int i = blockIdx.x * blockDim.x + threadIdx.x;
    if (i < total) h0[i] = h1[i] * norm[i >> 7];
}

// SpMM hop: one wave per edge, 32 lanes x float4 = 128 features.
// Gather hits L2 (25.6MB feature matrix resident); scatter uses native f32 atomics.
__global__ void spmm_kernel(const float* __restrict__ hin, float* __restrict__ hout,
                            const int* __restrict__ src, const int* __restrict__ dst, int nE) {
    int gt   = blockIdx.x * blockDim.x + threadIdx.x;
    int e    = gt >> 5;
    int lane = gt & 31;
    if (e >= nE) return;
    int s = src[e];
    int d = dst[e];
    const float4 v = ((const float4*)(hin + (size_t)s * IN_F))[lane];
    float* o = hout + (size_t)d * IN_F + lane * 4;
    unsafeAtomicAdd(o + 0, v.x);
    unsafeAtomicAdd(o + 1, v.y);
    unsafeAtomicAdd(o + 2, v.z);
    unsafeAtomicAdd(o + 3, v.w);
}

// ---------------------------------------------------------------------------
// out[M x 64] = H[M x 128] @ W^T[128 x 64] + bias, via V_WMMA_F32_16X16X4_F32.
// One wave computes one 16x16 tile of C; 4 waves per block cover all 4 N-tiles.
// A (16x4 f32): lanes 0-15 -> K=0,1 ; lanes 16-31 -> K=2,3 (2 VGPRs / lane).
// B (4x16 f32): VGPR0 = rows K=0 (lanes 0-15) / K=2 (lanes 16-31); VGPR1 = K=1/K=3.
// C/D (16x16 f32): VGPR v -> row v (lanes 0-15) and row v+8 (lanes 16-31).
__global__ void sgc_gemm_wmma(const float* __restrict__ H, const float* __restrict__ W,
                              const float* __restrict__ bias, float* __restrict__ out,
                              int M) {
    const int lane = threadIdx.x & 31;
    const int wv   = threadIdx.x >> 5;          // N-tile 0..3
    const int m0   = blockIdx.x << 4;
    const int n0   = wv << 4;
    const int lr   = lane & 15;
    const int khi  = (lane >> 4) << 1;          // 0 for lanes 0-15, 2 for lanes 16-31

    int arowIdx = m0 + lr;
    if (arowIdx >= M) arowIdx = M - 1;          // clamp loads; EXEC stays all-1s for WMMA
    const float* arow = H + (size_t)arowIdx * IN_F + khi;
    const float* brow = W + (size_t)(n0 + lr) * IN_F + khi;   // B[k][n] = W[n][k]

    v8f acc = {};
#pragma unroll
    for (int k = 0; k < IN_F; k += 4) {
        v2f a; a.x = arow[k]; a.y = arow[k + 1];
        v2f b; b.x = brow[k]; b.y = brow[k + 1];
        acc = __builtin_amdgcn_wmma_f32_16x16x4_f32(
            /*neg_a=*/false, a, /*neg_b=*/false, b,
            /*c_mod=*/(short)0, acc, /*reuse_a=*/false, /*reuse_b=*/false);
    }

    const float bb  = bias[n0 + lr];
    const int mrow  = m0 + ((lane >> 4) << 3);  // +8 for hi half-lanes
#pragma unroll
    for (int v = 0; v < 8; ++v) {
        int m = mrow + v;
        if (m < M) out[(size_t)m * OUT_F + (n0 + lr)] = acc[v] + bb;
    }
}

// ---------------------------------------------------------------------------
extern "C" void kernel_launch(void* const* d_in, const int* in_sizes, int n_in,
                              void* d_out, int out_size, void* d_ws, size_t ws_size,
                              hipStream_t stream) {
    const float* x   = (const float*)d_in[0];
    const float* W   = (const float*)d_in[1];
    const float* b   = (const float*)d_in[2];
    const int*   src = (const int*)d_in[3];
    const int*   dst = (const int*)d_in[4];

    const int nNodes = in_sizes[0] / IN_F;
    const int nE     = in_sizes[3];
    const int total  = nNodes * IN_F;

    float* norm = (float*)d_ws;
    float* h0   = norm + (((size_t)nNodes + 63) & ~(size_t)63);
    float* h1   = h0 + (size_t)nNodes * IN_F;

    const int B = 256;
    const int gNode  = (nNodes + B - 1) / B;
    const int gEdge  = (nE + B - 1) / B;
    const int gFeat  = (total + B - 1) / B;
    const long eth   = (long)nE * 32;
    const int gSpmm  = (int)((eth + B - 1) / B);
    const int gGemm  = (nNodes + 15) / 16;

    // degrees -> norm
    zero_f32<<<gNode, B, 0, stream>>>(norm, nNodes);
    degree_kernel<<<gEdge, B, 0, stream>>>(dst, norm, nE);
    norm_kernel<<<gNode, B, 0, stream>>>(norm, nNodes);

    // hop 1
    prescale_kernel<<<gFeat, B, 0, stream>>>(x, norm, h0, h1, total);
    spmm_kernel<<<gSpmm, B, 0, stream>>>(h0, h1, src, dst, nE);

    // hop 2 (fused dst-scale * src-scale = norm^2)
    midscale_kernel<<<gFeat, B, 0, stream>>>(norm, h0, h1, total);
    spmm_kernel<<<gSpmm, B, 0, stream>>>(h0, h1, src, dst, nE);

    // final dst-scale, then fc via WMMA
    finalscale_kernel<<<gFeat, B, 0, stream>>>(norm, h0, h1, total);
    sgc_gemm_wmma<<<gGemm, 128, 0, stream>>>(h0, W, b, (float*)d_out, nNodes);
}